// Attn_128849019074
// MI455X (gfx1250) — compile-verified
//
#include <hip/hip_runtime.h>

typedef __attribute__((ext_vector_type(2))) float v2f;
typedef __attribute__((ext_vector_type(4))) float v4f;
typedef __attribute__((ext_vector_type(8))) float v8f;

#define HID   1024
#define DD    2048      // 2*HID
#define LL    1024
#define BB    32

// ---------------------------------------------------------------------------
// Kernel 1: V = hidden(32x2048) @ W(2048x2048), fp32 via V_WMMA_F32_16X16X4_F32
// Each wave owns one (16-row M-tile, 16-col N-tile) pair, full K.
// 2 M-tiles * 128 N-tiles = 256 waves = 32 blocks * 8 waves.
//
// A (16x4 fp32, 2 VGPRs/lane): lanes 0-15 rows 0-15 hold K={k0,k0+1};
//                              lanes 16-31 hold K={k0+2,k0+3}. (ISA 7.12.2)
// B (4x16 fp32, 2 VGPRs/lane): mirrored: lanes 0-15 col n hold rows {k0,k0+1};
//                              lanes 16-31 hold rows {k0+2,k0+3}.
// C/D (16x16 fp32, 8 VGPRs):   VGPR r -> row r (lanes 0-15) / row r+8 (16-31).
// ---------------------------------------------------------------------------
__global__ __launch_bounds__(256)
void hw_gemm_wmma(const float* __restrict__ hidden,
                  const float* __restrict__ W,
                  float* __restrict__ vbuf) {
  const int lane = threadIdx.x & 31;
  const int wid  = (blockIdx.x * blockDim.x + threadIdx.x) >> 5;  // 0..255
  const int m0   = (wid & 1) * 16;        // row tile of hidden
  const int n0   = (wid >> 1) * 16;       // col tile of W
  const int half = lane >> 4;             // 0: lanes 0-15, 1: lanes 16-31
  const int l15  = lane & 15;

  const float* Arow = hidden + (m0 + l15) * DD;  // this lane's A row
  const int nc = n0 + l15;                        // this lane's B/N column

  v8f acc0 = {};
  v8f acc1 = {};
#pragma unroll 2
  for (int k0 = 0; k0 < DD; k0 += 8) {
    // --- k step A (K = k0..k0+3) ---
    const int ka = k0 + 2 * half;                  // even -> 8B aligned
    v2f aA = *(const v2f*)(Arow + ka);
    v2f bA;
    bA[0] = W[(size_t)ka * DD + nc];
    bA[1] = W[(size_t)(ka + 1) * DD + nc];
    acc0 = __builtin_amdgcn_wmma_f32_16x16x4_f32(
        false, aA, false, bA, (short)0, acc0, false, false);

    // --- k step B (K = k0+4..k0+7), independent accumulator ---
    const int kb = k0 + 4 + 2 * half;
    v2f aB = *(const v2f*)(Arow + kb);
    v2f bB;
    bB[0] = W[(size_t)kb * DD + nc];
    bB[1] = W[(size_t)(kb + 1) * DD + nc];
    acc1 = __builtin_amdgcn_wmma_f32_16x16x4_f32(
        false, aB, false, bB, (short)0, acc1, false, false);
  }
  v8f acc = acc0 + acc1;

  const int rowbase = m0 + 8 * half;
#pragma unroll
  for (int r = 0; r < 8; ++r)
    vbuf[(size_t)(rowbase + r) * DD + nc] = acc[r];
}

// ---------------------------------------------------------------------------
// Kernel 1b: c[b] = bias . hidden[b,:]   (tiny)
// ---------------------------------------------------------------------------
__global__ __launch_bounds__(256)
void bias_dot(const float* __restrict__ bias,
              const float* __restrict__ hidden,
              float* __restrict__ cvec) {
  const int b = blockIdx.x;
  float sum = 0.f;
  for (int i = threadIdx.x; i < DD; i += 256)
    sum += bias[i] * hidden[(size_t)b * DD + i];
#pragma unroll
  for (int off = 16; off; off >>= 1) sum += __shfl_xor(sum, off, 32);
  __shared__ float red[8];
  const int lane = threadIdx.x & 31, wave = threadIdx.x >> 5;
  if (lane == 0) red[wave] = sum;
  __syncthreads();
  if (threadIdx.x < 8) {
    float s = red[threadIdx.x];
#pragma unroll
    for (int off = 4; off; off >>= 1) s += __shfl_xor(s, off, 32);
    if (threadIdx.x == 0) cvec[b] = s;
  }
}

// ---------------------------------------------------------------------------
// Kernel 2 (the 268 MB streaming pass): one block per l.
// scores[l,b] = enc[l,b,:] . v[b,:] + c[b]; then log_softmax over b (32-wide).
// 8 waves/block, each wave reduces 4 batch rows with coalesced float4 loads.
// enc is single-use -> non-temporal loads keep v/W resident in L2.
// ---------------------------------------------------------------------------
__global__ __launch_bounds__(256)
void scores_logsoftmax(const float* __restrict__ enc,
                       const float* __restrict__ vbuf,
                       const float* __restrict__ cvec,
                       float* __restrict__ out) {
  const int l    = blockIdx.x;
  const int lane = threadIdx.x & 31;
  const int wave = threadIdx.x >> 5;  // 0..7

  __shared__ float s_sc[BB];
  const float* encL = enc + (size_t)l * BB * DD;

#pragma unroll
  for (int bb = 0; bb < 4; ++bb) {
    const int b = wave * 4 + bb;
    const float* e = encL + (size_t)b * DD;
    const float* v = vbuf + (size_t)b * DD;
    float sum = 0.f;
#pragma unroll 4
    for (int it = 0; it < 16; ++it) {
      const int idx = (it * 32 + lane) * 4;       // contiguous 512B per wave
      v4f ev = __builtin_nontemporal_load((const v4f*)(e + idx));
      v4f vv = *(const v4f*)(v + idx);
      sum = fmaf(ev[0], vv[0], sum);
      sum = fmaf(ev[1], vv[1], sum);
      sum = fmaf(ev[2], vv[2], sum);
      sum = fmaf(ev[3], vv[3], sum);
    }
#pragma unroll
    for (int off = 16; off; off >>= 1) sum += __shfl_xor(sum, off, 32);
    if (lane == 0) s_sc[b] = sum + cvec[b];
  }
  __syncthreads();

  // 32-wide log-softmax over the batch dim, done by wave 0
  if (threadIdx.x < 32) {
    const float s = s_sc[lane];
    float m = s;
#pragma unroll
    for (int off = 16; off; off >>= 1) m = fmaxf(m, __shfl_xor(m, off, 32));
    float ex = __expf(s - m);
    float tot = ex;
#pragma unroll
    for (int off = 16; off; off >>= 1) tot += __shfl_xor(tot, off, 32);
    out[(size_t)l * BB + lane] = s - m - __logf(tot);
  }
}

// ---------------------------------------------------------------------------
extern "C" void kernel_launch(void* const* d_in, const int* in_sizes, int n_in,
                              void* d_out, int out_size, void* d_ws, size_t ws_size,
                              hipStream_t stream) {
  const float* hidden = (const float*)d_in[0];   // [B, D]
  const float* enc    = (const float*)d_in[1];   // [L, B, D]
  const float* W      = (const float*)d_in[2];   // [D, D]
  const float* bias   = (const float*)d_in[3];   // [D]
  float* out  = (float*)d_out;                   // [L*B]
  float* vbuf = (float*)d_ws;                    // 32*2048 floats = 256 KB
  float* cvec = vbuf + BB * DD;                  // 32 floats

  hw_gemm_wmma<<<dim3(32), dim3(256), 0, stream>>>(hidden, W, vbuf);
  bias_dot<<<dim3(BB), dim3(256), 0, stream>>>(bias, hidden, cvec);
  scores_logsoftmax<<<dim3(LL), dim3(256), 0, stream>>>(enc, vbuf, cvec, out);
}